// LsqWeight_3b_4595615007451
// MI455X (gfx1250) — compile-verified
//
#include <hip/hip_runtime.h>
#include <cmath>
#include <cstdint>

#define TPB   256
#define CHUNK (TPB * 4)   // 1024 floats = 4 KB per chunk, double-buffered in LDS

typedef __attribute__((ext_vector_type(4))) int v4i;

// ---- exact reproduction of _grad_scale value: (s - s*g) + s*g, fp32 ----
// pragma prevents FMA contraction (reference/XLA rounds mul, sub, add separately;
// round-2 asm showed hipcc fusing these into s_fmac_f32 otherwise)
__device__ __forceinline__ float grad_scale_val(float s, float g) {
#pragma clang fp contract(off)
    float m = s * g;
    float v = s - m;
    return v + m;
}

// ---- CDNA5 async global -> LDS copy (ASYNCcnt path) ----
__device__ __forceinline__ void async_load_b128(const float* gp, float* lp) {
#if __has_builtin(__builtin_amdgcn_global_load_async_to_lds_b128)
    // signature (from clang diagnostic): (int4 AS1*, int4 AS3*, imm offset, imm cpol)
    __builtin_amdgcn_global_load_async_to_lds_b128(
        (__attribute__((address_space(1))) v4i*)(v4i*)gp,
        (__attribute__((address_space(3))) v4i*)(v4i*)lp,
        0, 0);
#else
    // ISA: GLOBAL_LOAD_ASYNC_TO_LDS_B128, GV mode: copies 16 B from global VADDR to the LDS offset in VDST
    unsigned loff = (unsigned)(uintptr_t)lp;          // low 32 bits of generic LDS addr = LDS offset
    unsigned long long ga = (unsigned long long)(uintptr_t)gp;
    asm volatile("global_load_async_to_lds_b128 %0, %1, off"
                 : : "v"(loff), "v"(ga) : "memory");
#endif
}

template <int N>
__device__ __forceinline__ void wait_async() {
#if __has_builtin(__builtin_amdgcn_s_wait_asynccnt)
    __builtin_amdgcn_s_wait_asynccnt(N);
#else
    if constexpr (N == 0) { asm volatile("s_wait_asynccnt 0x0" ::: "memory"); }
    else                  { asm volatile("s_wait_asynccnt 0x1" ::: "memory"); }
#endif
}

// ---- branch-free nearest-of-27 (first index wins ties, like jnp.argmin) ----
__device__ __forceinline__ float quant27(const float lv[27], float x) {
    float best = lv[0];
    float dmin = __builtin_fabsf(x - lv[0]);
#pragma unroll
    for (int k = 1; k < 27; ++k) {
        float d = __builtin_fabsf(x - lv[k]);
        if (d < dmin) { dmin = d; best = lv[k]; }   // strict < keeps earliest k on tie
    }
    return best;
}

__global__ __launch_bounds__(TPB) void lsq3b_quant_kernel(
    const float* __restrict__ x,  const float* __restrict__ sH,
    const float* __restrict__ sM, const float* __restrict__ sL,
    float* __restrict__ out, int C, int N, float g)
{
    __shared__ float smem[2][CHUNK];

    const int c = blockIdx.x;
    const int t = threadIdx.x;

    // per-channel perturbed scales (uniform across block -> scalar loads)
    const float hs = grad_scale_val(sH[c], g);
    const float ms = grad_scale_val(sM[c], g);
    const float ls = grad_scale_val(sL[c], g);

    if (t == 0) {
        const size_t base = (size_t)C * (size_t)N;
        out[base + c]     = hs;   // sHs
        out[base + C + c] = ls;   // sLs
    }

    // 27-entry codebook, cH-major (meshgrid 'ij'): k = 9*i + 3*j + l
    // products by {-1,0,1} are exact; adds in reference's left-to-right order
    float lv[27];
    {
#pragma clang fp contract(off)
#pragma unroll
        for (int i = 0; i < 3; ++i)
#pragma unroll
            for (int j = 0; j < 3; ++j)
#pragma unroll
                for (int l = 0; l < 3; ++l) {
                    const float a = hs * (float)(i - 1);
                    const float b = ms * (float)(j - 1);
                    const float d = ls * (float)(l - 1);
                    lv[9 * i + 3 * j + l] = (a + b) + d;
                }
    }

    const float* xc = x   + (size_t)c * (size_t)N;
    float*       oc = out + (size_t)c * (size_t)N;

    const int nch = N / CHUNK;

    if (nch > 0) {
        // prologue: stage chunk 0
        async_load_b128(xc + 4 * t, &smem[0][4 * t]);

        for (int ch = 0; ch < nch; ++ch) {
            const int cur = ch & 1;
            if (ch + 1 < nch) {
                // prefetch next chunk into the other buffer, then wait for current
                async_load_b128(xc + (size_t)(ch + 1) * CHUNK + 4 * t,
                                &smem[cur ^ 1][4 * t]);
                wait_async<1>();   // async loads complete in order: current chunk done
            } else {
                wait_async<0>();
            }
            __syncthreads();       // all waves' portion of chunk `ch` now in LDS

            float4 v = *reinterpret_cast<const float4*>(&smem[cur][4 * t]);
            float4 r;
            r.x = quant27(lv, v.x);
            r.y = quant27(lv, v.y);
            r.z = quant27(lv, v.z);
            r.w = quant27(lv, v.w);
            *reinterpret_cast<float4*>(oc + (size_t)ch * CHUNK + 4 * t) = r;

            __syncthreads();       // protect buf[cur] from being re-filled at iter ch+1
        }
    }

    // tail (N not a multiple of CHUNK) — direct global path
    for (int i = nch * CHUNK + t; i < N; i += TPB)
        oc[i] = quant27(lv, xc[i]);
}

extern "C" void kernel_launch(void* const* d_in, const int* in_sizes, int n_in,
                              void* d_out, int out_size, void* d_ws, size_t ws_size,
                              hipStream_t stream)
{
    const float* x  = (const float*)d_in[0];
    const float* sH = (const float*)d_in[1];
    const float* sM = (const float*)d_in[2];
    const float* sL = (const float*)d_in[3];
    float* out = (float*)d_out;

    const int C = in_sizes[1];            // 1024
    const int N = in_sizes[0] / C;        // 4096

    // g = 1/sqrt(THD_POS * n_per_ch)/10 in fp32, same op order as reference
    const float g = 1.0f / sqrtf((float)(7 * N)) / 10.0f;

    lsq3b_quant_kernel<<<dim3(C), dim3(TPB), 0, stream>>>(x, sH, sM, sL, out, C, N, g);
}